// P_Pooling_Module_69784628625876
// MI455X (gfx1250) — compile-verified
//
#include <hip/hip_runtime.h>

// Problem constants (from reference)
#define BB 16
#define CC 256
#define HH 128
#define WW 128
#define OHT 64
#define OWT 64
#define TILE_OH 16                 // output rows per block
#define TROWS (2 * TILE_OH + 1)    // 33 input rows incl. halo
#define ROW_STRIDE 136             // floats per LDS row: [3]=iw -1, [4..131]=iw 0..127, [132]=iw 128
#define NTHREADS 256

#if __has_builtin(__builtin_amdgcn_global_load_async_to_lds_b128)
#define HAVE_ASYNC_B128 1
#else
#define HAVE_ASYNC_B128 0
#endif

typedef int v4i __attribute__((ext_vector_type(4)));
typedef __attribute__((address_space(1))) v4i* g_v4i_ptr;   // global (AS1)
typedef __attribute__((address_space(3))) v4i* l_v4i_ptr;   // LDS (AS3)

static __device__ __forceinline__ float fast_log2(float x) {
  return __builtin_amdgcn_logf(x);    // v_log_f32 (log2); log2(0) = -inf
}
static __device__ __forceinline__ float fast_exp2(float x) {
  return __builtin_amdgcn_exp2f(x);   // v_exp_f32 (exp2); exp2(-inf) = 0
}

static __device__ __forceinline__ void wait_async0() {
#if __has_builtin(__builtin_amdgcn_s_wait_asynccnt)
  __builtin_amdgcn_s_wait_asynccnt(0);
#else
  asm volatile("s_wait_asynccnt 0" ::: "memory");
#endif
}

__global__ __launch_bounds__(NTHREADS) void p_pool_fused_kernel(
    const float* __restrict__ bottom,
    const float* __restrict__ w1,
    const float* __restrict__ alpha,
    const float* __restrict__ w2,
    const float* __restrict__ b2,
    float* __restrict__ out) {
  __shared__ float tile [TROWS * ROW_STRIDE];   // raw input values
  __shared__ float ltile[TROWS * ROW_STRIDE];   // log2 of input values (shared across overlapping windows)

  const int t   = threadIdx.x;
  const int tb  = blockIdx.x;   // 0..3 row tile
  const int ch  = blockIdx.y;   // channel
  const int b   = blockIdx.z;   // batch

  const int oh0 = tb * TILE_OH;
  const int ih0 = 2 * oh0 - 1;  // first input row of the tile (may be -1)

  const float* plane = bottom + (size_t)(b * CC + ch) * (HH * WW);

  // Uniform (per-block) channel parameters -> scalar loads
  float k[9];
#pragma unroll
  for (int i = 0; i < 9; ++i) k[i] = w1[ch * 9 + i];
  const float a  = alpha[ch];
  const float s2 = w2[ch];
  const float c2 = b2[ch];

  // ---- Zero-fill halo columns and OOB rows (disjoint from async interior) ----
  if (t < TROWS) {
    tile[t * ROW_STRIDE + 3]   = 0.0f;  // iw = -1
    tile[t * ROW_STRIDE + 132] = 0.0f;  // iw = 128
  }
  if (ih0 < 0) {  // first tile: tile row 0 is input row -1
    for (int j = t; j < WW; j += NTHREADS) tile[0 * ROW_STRIDE + 4 + j] = 0.0f;
  }
  if (ih0 + TROWS - 1 >= HH) {  // last tile: tile row 32 is input row 127 (in range)
    for (int j = t; j < WW; j += NTHREADS) tile[(TROWS - 1) * ROW_STRIDE + 4 + j] = 0.0f;
  }

  // ---- Stage interior rows: global -> LDS via async b128 copies (ASYNCcnt path) ----
  const int r_lo = (ih0 < 0) ? 1 : 0;
  const int r_hi = (ih0 + TROWS - 1 >= HH) ? (TROWS - 2) : (TROWS - 1);
  const int nchunks = (r_hi - r_lo + 1) * (WW / 4);  // float4 chunks
  for (int c = t; c < nchunks; c += NTHREADS) {
    const int r  = r_lo + (c >> 5);      // tile row
    const int j4 = (c & 31) * 4;         // starting column (multiple of 4)
    const int ih = ih0 + r;
    const float* g = plane + (size_t)ih * WW + j4;
    float* l = &tile[r * ROW_STRIDE + 4 + j4];  // 16B aligned
#if HAVE_ASYNC_B128
    __builtin_amdgcn_global_load_async_to_lds_b128((g_v4i_ptr)g, (l_v4i_ptr)l, 0, 0);
#else
    *(float4*)l = *(const float4*)g;
#endif
  }
#if HAVE_ASYNC_B128
  wait_async0();
#endif
  __syncthreads();

  // ---- Precompute log2 of the tile once (each input pixel feeds ~2.25 windows) ----
  for (int j = t; j < TROWS * ROW_STRIDE; j += NTHREADS) {
    ltile[j] = fast_log2(tile[j]);
  }
  __syncthreads();

  // ---- Compute: each thread produces 4 output pixels ----
  const int tx = t & 63;   // ow
  const int ty = t >> 6;   // 0..3
  float* oplane = out + (size_t)(b * CC + ch) * (OHT * OWT);

#pragma unroll
  for (int q = 0; q < 4; ++q) {
    const int ohh = ty + 4 * q;        // 0..15 within tile
    const int lr  = 2 * ohh;           // top row of 3x3 patch in tile
    const int lc  = 3 + 2 * tx;        // left col of patch (iw = 2*ow - 1)

    const int i0 = (lr + 0) * ROW_STRIDE + lc;
    const int i1 = (lr + 1) * ROW_STRIDE + lc;
    const int i2 = (lr + 2) * ROW_STRIDE + lc;

    float v[9], lv[9];
    v[0] = tile[i0];  v[1] = tile[i0 + 1];  v[2] = tile[i0 + 2];
    v[3] = tile[i1];  v[4] = tile[i1 + 1];  v[5] = tile[i1 + 2];
    v[6] = tile[i2];  v[7] = tile[i2 + 1];  v[8] = tile[i2 + 2];
    lv[0] = ltile[i0]; lv[1] = ltile[i0 + 1]; lv[2] = ltile[i0 + 2];
    lv[3] = ltile[i1]; lv[4] = ltile[i1 + 1]; lv[5] = ltile[i1 + 2];
    lv[6] = ltile[i2]; lv[7] = ltile[i2 + 1]; lv[8] = ltile[i2 + 2];

    // Depthwise 3x3 conv (SCALE == 1)
    float x = 0.0f;
#pragma unroll
    for (int i = 0; i < 9; ++i) x = fmaf(v[i], k[i], x);
    // PReLU
    x = (x >= 0.0f) ? x : a * x;
    // Affine + clip -> per-pixel exponent p in [1, 50]
    const float p = fminf(fmaxf(fmaf(x, s2, c2), 1.0f), 50.0f);

    // p-pooling: (sum v^p)^(1/p); padded taps: exp2(p * -inf) = 0
    float acc = 0.0f;
#pragma unroll
    for (int i = 0; i < 9; ++i) acc += fast_exp2(p * lv[i]);
    const float res = fast_exp2(fast_log2(acc) / p);

    oplane[(size_t)(oh0 + ohh) * OWT + tx] = res;
  }
}

extern "C" void kernel_launch(void* const* d_in, const int* in_sizes, int n_in,
                              void* d_out, int out_size, void* d_ws, size_t ws_size,
                              hipStream_t stream) {
  const float* bottom = (const float*)d_in[0];
  const float* w1     = (const float*)d_in[1];
  const float* alpha  = (const float*)d_in[2];
  const float* w2     = (const float*)d_in[3];
  const float* b2     = (const float*)d_in[4];
  float* out = (float*)d_out;

  dim3 grid(OHT / TILE_OH, CC, BB);   // (4, 256, 16) blocks
  p_pool_fused_kernel<<<grid, NTHREADS, 0, stream>>>(bottom, w1, alpha, w2, b2, out);
}